// PANM_10900626997733
// MI455X (gfx1250) — compile-verified
//
// PANM seq2seq forward for MI455X (gfx1250), fp32 via V_WMMA_F32_16X16X4_F32.
//
// Strategy (see analysis): WMMA-f32 tiled GEMM for all weight matmuls;
// step-invariant hoists (Kp/Vp/Kc/Vc and the Bahdanau enc@W2^T term) done once
// as large GEMMs; HBM-bound attention reductions as coalesced wave32 kernels.

#include <hip/hip_runtime.h>
#include <math.h>

// ---------------- model dims ----------------
constexpr int S_  = 256;
constexpr int B_  = 128;
constexpr int E_  = 512;
constexpr int H_  = 512;
constexpr int ADD_= 8;
constexpr int HA_ = 2;
constexpr int T_  = 64;
constexpr int OUT_= 512;
constexpr int G_  = 3 * H_;              // 1536
constexpr int DIN_= E_ + HA_ * H_ + H_;  // 2048

typedef float v2f __attribute__((ext_vector_type(2)));
typedef float v8f __attribute__((ext_vector_type(8)));

// =====================================================================
// WMMA f32 GEMM:  C(M,N) = A(M,K) @ W(N,K)^T  [+ bias(N)]
// Row-major, leading dims lda/ldw/ldc. Requires M%64==0, N%256==0
// (true for every call here: M in {128, 32768}, N in {512, 1536}).
// K arbitrary (zero-padded in LDS), covers K=8 pointer-unit GEMMs.
// Block: 256 thr = 8 waves (2x4); wave tile 32x64; block tile 64x256.
// =====================================================================
__global__ __launch_bounds__(256) void gemm_f32_wmma(
    const float* __restrict__ A, int lda,
    const float* __restrict__ W, int ldw,
    const float* __restrict__ bias,
    float* __restrict__ C, int ldc,
    int M, int N, int K)
{
  __shared__ float lA[64][17];   // +1 pad: conflict-free frag reads
  __shared__ float lB[256][17];

  const int tid  = threadIdx.x;
  const int m0   = blockIdx.y * 64;
  const int n0   = blockIdx.x * 256;
  const int wave = tid >> 5;
  const int lane = tid & 31;
  const int wm   = wave >> 2;          // 0..1
  const int wn   = wave & 3;           // 0..3
  const int l    = lane & 15;
  const int kh   = (lane >> 4) * 2;    // 0 or 2 (K-half per WMMA layout)

  const v8f vzero = {0.f,0.f,0.f,0.f,0.f,0.f,0.f,0.f};
  v8f acc[2][4];
  #pragma unroll
  for (int i = 0; i < 2; ++i)
    #pragma unroll
    for (int j = 0; j < 4; ++j) acc[i][j] = vzero;

  const int arow = tid >> 2;           // 0..63
  const int akq  = (tid & 3) * 4;      // 0,4,8,12

  for (int kt = 0; kt < K; kt += 16) {
    // ---- stage A tile (64 x 16), 4 floats/thread ----
    {
      const float* ap = A + (size_t)(m0 + arow) * lda + kt + akq;
      if (kt + akq + 3 < K) {
        lA[arow][akq+0] = ap[0]; lA[arow][akq+1] = ap[1];
        lA[arow][akq+2] = ap[2]; lA[arow][akq+3] = ap[3];
      } else {
        #pragma unroll
        for (int q = 0; q < 4; ++q)
          lA[arow][akq+q] = (kt + akq + q < K) ? ap[q] : 0.0f;
      }
    }
    // ---- stage B tile (256 x 16), 16 floats/thread ----
    #pragma unroll
    for (int i = 0; i < 4; ++i) {
      const int nrow = (tid >> 2) + i * 64;
      const float* wp = W + (size_t)(n0 + nrow) * ldw + kt + akq;
      if (kt + akq + 3 < K) {
        lB[nrow][akq+0] = wp[0]; lB[nrow][akq+1] = wp[1];
        lB[nrow][akq+2] = wp[2]; lB[nrow][akq+3] = wp[3];
      } else {
        #pragma unroll
        for (int q = 0; q < 4; ++q)
          lB[nrow][akq+q] = (kt + akq + q < K) ? wp[q] : 0.0f;
      }
    }
    __syncthreads();

    // ---- 4 k-steps of 4 => 32 v_wmma_f32_16x16x4_f32 per stage ----
    #pragma unroll
    for (int kk = 0; kk < 4; ++kk) {
      const int kb = kk * 4 + kh;
      v2f afrag[2], bfrag[4];
      #pragma unroll
      for (int i = 0; i < 2; ++i) {
        const int r = wm * 32 + i * 16 + l;
        afrag[i].x = lA[r][kb];
        afrag[i].y = lA[r][kb + 1];
      }
      #pragma unroll
      for (int j = 0; j < 4; ++j) {
        const int c = wn * 64 + j * 16 + l;
        bfrag[j].x = lB[c][kb];
        bfrag[j].y = lB[c][kb + 1];
      }
      #pragma unroll
      for (int i = 0; i < 2; ++i)
        #pragma unroll
        for (int j = 0; j < 4; ++j)
          acc[i][j] = __builtin_amdgcn_wmma_f32_16x16x4_f32(
              false, afrag[i], false, bfrag[j], (short)0, acc[i][j],
              false, false);
    }
    __syncthreads();
  }

  // ---- store C tile (+ bias) per 16x16 f32 D layout ----
  const int rbase = (lane < 16) ? 0 : 8;
  #pragma unroll
  for (int i = 0; i < 2; ++i) {
    #pragma unroll
    for (int j = 0; j < 4; ++j) {
      const int ccol = n0 + wn * 64 + j * 16 + l;
      const float bv = bias ? bias[ccol] : 0.0f;
      #pragma unroll
      for (int v = 0; v < 8; ++v) {
        const int crow = m0 + wm * 32 + i * 16 + rbase + v;
        C[(size_t)crow * ldc + ccol] = acc[i][j][v] + bv;
      }
    }
  }
}

// =====================================================================
// wave32 helpers
// =====================================================================
__device__ __forceinline__ float waveReduceSum(float v) {
  #pragma unroll
  for (int o = 16; o > 0; o >>= 1) v += __shfl_xor(v, o, 32);
  return v;
}

// GRU gate fusion: reads gi/gh (B,G) with biases already applied by GEMM,
// updates h (B,H) in place; optionally writes/accumulates into enc slot.
__global__ __launch_bounds__(256) void gru_gate_kernel(
    const float* __restrict__ gi, const float* __restrict__ gh,
    float* __restrict__ h, float* __restrict__ encslot, int addmode)
{
  const int idx = blockIdx.x * 256 + threadIdx.x;  // over B*H
  const int b = idx >> 9, j = idx & 511;
  const size_t o = (size_t)b * G_ + j;
  const float r = 1.0f / (1.0f + __expf(-(gi[o] + gh[o])));
  const float z = 1.0f / (1.0f + __expf(-(gi[o + H_] + gh[o + H_])));
  const float n = tanhf(gi[o + 2 * H_] + r * gh[o + 2 * H_]);
  const float h2 = (1.0f - z) * n + z * h[idx];
  h[idx] = h2;
  if (encslot) { if (addmode) encslot[idx] += h2; else encslot[idx] = h2; }
}

// sc[b,s] = scale * sum_h q[b,h] * Kmat[s,b,h]   (one wave per (b,s))
__global__ __launch_bounds__(256) void dot_scores_kernel(
    const float* __restrict__ q, const float* __restrict__ Kmat,
    float* __restrict__ sc, float scale)
{
  const int gw = (blockIdx.x * 256 + threadIdx.x) >> 5;  // b*S + s
  const int lane = threadIdx.x & 31;
  const int b = gw >> 8, s = gw & 255;
  const float* qp = q + (size_t)b * H_;
  const float* kp = Kmat + ((size_t)s * B_ + b) * H_;
  float acc = 0.0f;
  for (int hh = lane; hh < H_; hh += 32) acc += qp[hh] * kp[hh];
  acc = waveReduceSum(acc);
  if (lane == 0) sc[b * S_ + s] = acc * scale;
}

// Bahdanau: sc[b,s] = sum_h relu(e1[b,h] + E2[s,b,h]) * v[h]
__global__ __launch_bounds__(256) void bahdanau_scores_kernel(
    const float* __restrict__ e1, const float* __restrict__ E2mat,
    const float* __restrict__ vvec, float* __restrict__ sc)
{
  const int gw = (blockIdx.x * 256 + threadIdx.x) >> 5;
  const int lane = threadIdx.x & 31;
  const int b = gw >> 8, s = gw & 255;
  const float* ep = e1 + (size_t)b * H_;
  const float* Ep = E2mat + ((size_t)s * B_ + b) * H_;
  float acc = 0.0f;
  for (int hh = lane; hh < H_; hh += 32)
    acc += fmaxf(ep[hh] + Ep[hh], 0.0f) * vvec[hh];
  acc = waveReduceSum(acc);
  if (lane == 0) sc[b * S_ + s] = acc;
}

// in-place softmax over rows of length 256 (one block per row)
__global__ __launch_bounds__(256) void softmax256_kernel(float* __restrict__ p)
{
  __shared__ float red[8];
  const int r = blockIdx.x, t = threadIdx.x;
  const float v = p[(size_t)r * 256 + t];
  float m = v;
  #pragma unroll
  for (int o = 16; o > 0; o >>= 1) m = fmaxf(m, __shfl_xor(m, o, 32));
  if ((t & 31) == 0) red[t >> 5] = m;
  __syncthreads();
  float bm = red[0];
  #pragma unroll
  for (int i = 1; i < 8; ++i) bm = fmaxf(bm, red[i]);
  const float e = __expf(v - bm);
  float sdot = waveReduceSum(e);
  __syncthreads();
  if ((t & 31) == 0) red[t >> 5] = sdot;
  __syncthreads();
  float tot = 0.0f;
  #pragma unroll
  for (int i = 0; i < 8; ++i) tot += red[i];
  p[(size_t)r * 256 + t] = e / tot;
}

// out[b,h] = sum_s aw[b,s] * V[s,b,h]    (coalesced over h; aw row in LDS)
__global__ __launch_bounds__(256) void attn_apply_kernel(
    const float* __restrict__ aw, const float* __restrict__ V,
    float* __restrict__ out)
{
  __shared__ float sa[256];
  const int blk = blockIdx.x;           // 2 blocks per batch row
  const int b = blk >> 1, h0 = (blk & 1) * 256;
  sa[threadIdx.x] = aw[b * S_ + threadIdx.x];
  __syncthreads();
  const int h = h0 + threadIdx.x;
  const float* vp = V + (size_t)b * H_ + h;
  float acc = 0.0f;
  for (int s = 0; s < S_; ++s) acc += sa[s] * vp[(size_t)s * B_ * H_];
  out[(size_t)b * H_ + h] = acc;
}

// ptrs_u[b,a] = sum_s aw[b,s] * A[s,b,a]
__global__ __launch_bounds__(256) void cadd_kernel(
    const float* __restrict__ aw, const float* __restrict__ Aaddr,
    float* __restrict__ dst)
{
  const int idx = blockIdx.x * 256 + threadIdx.x;   // B*ADD = 1024
  if (idx >= B_ * ADD_) return;
  const int b = idx >> 3, a = idx & 7;
  float acc = 0.0f;
  for (int s = 0; s < S_; ++s)
    acc += aw[b * S_ + s] * Aaddr[((size_t)s * B_ + b) * ADD_ + a];
  dst[idx] = acc;
}

// Am[u,s,b,c] = sum_a A[s,b,a]*AmapW[u,c,a] + Amapb[u,c]
__global__ __launch_bounds__(256) void amap_kernel(
    const float* __restrict__ A, const float* __restrict__ Wm,
    const float* __restrict__ bm, float* __restrict__ Am)
{
  const int idx = blockIdx.x * 256 + threadIdx.x;   // HA*S*B*ADD
  const int a = idx & 7;
  int rest = idx >> 3;
  const int b = rest & 127; rest >>= 7;
  const int s = rest & 255;
  const int u = rest >> 8;
  const float* ap = A + ((size_t)s * B_ + b) * ADD_;
  const float* wp = Wm + u * ADD_ * ADD_ + a * ADD_;
  float acc = bm[u * ADD_ + a];
  #pragma unroll
  for (int a2 = 0; a2 < ADD_; ++a2) acc += ap[a2] * wp[a2];
  Am[idx] = acc;
}

// ptrs[0]=A[0], ptrs[1]=A[S-1]
__global__ __launch_bounds__(256) void init_ptrs_kernel(
    const float* __restrict__ A, float* __restrict__ ptrs)
{
  const int idx = blockIdx.x * 256 + threadIdx.x;   // HA*B*ADD = 2048
  if (idx >= HA_ * B_ * ADD_) return;
  const int u = idx >> 10, r = idx & 1023;
  ptrs[idx] = A[(u ? (size_t)(S_ - 1) * B_ * ADD_ : 0) + r];
}

// dst[r, c0..c0+cols) = src[r, 0..cols) with leading dims
__global__ __launch_bounds__(256) void copy_cols_kernel(
    const float* __restrict__ src, int lds, float* __restrict__ dst, int ldd,
    int cols)
{
  const int idx = blockIdx.x * 256 + threadIdx.x;   // B*cols
  const int r = idx / cols, c = idx % cols;
  if (r < B_) dst[(size_t)r * ldd + c] = src[(size_t)r * lds + c];
}

__global__ __launch_bounds__(256) void zero_kernel(float* __restrict__ p, int n)
{
  const int i = blockIdx.x * 256 + threadIdx.x;
  if (i < n) p[i] = 0.0f;
}

// =====================================================================
// host orchestration
// =====================================================================
static inline void gemm(hipStream_t st, const float* A, int lda,
                        const float* W, int ldw, const float* bias,
                        float* C, int ldc, int M, int N, int K)
{
  dim3 grid(N / 256, M / 64);
  gemm_f32_wmma<<<grid, 256, 0, st>>>(A, lda, W, ldw, bias, C, ldc, M, N, K);
}

extern "C" void kernel_launch(void* const* d_in, const int* in_sizes, int n_in,
                              void* d_out, int out_size, void* d_ws,
                              size_t ws_size, hipStream_t stream)
{
  (void)in_sizes; (void)n_in; (void)out_size; (void)ws_size;

  const float* src        = (const float*)d_in[0];
  const float* tgt        = (const float*)d_in[1];
  const float* Aaddr      = (const float*)d_in[2];
  const float* enc_Wih_f  = (const float*)d_in[3];
  const float* enc_Whh_f  = (const float*)d_in[4];
  const float* enc_bih_f  = (const float*)d_in[5];
  const float* enc_bhh_f  = (const float*)d_in[6];
  const float* enc_Wih_b  = (const float*)d_in[7];
  const float* enc_Whh_b  = (const float*)d_in[8];
  const float* enc_bih_b  = (const float*)d_in[9];
  const float* enc_bhh_b  = (const float*)d_in[10];
  const float* dec_attn_W = (const float*)d_in[11];
  const float* dec_attn_b = (const float*)d_in[12];
  const float* dec_attn_v = (const float*)d_in[13];
  const float* dec_Wih    = (const float*)d_in[14];
  const float* dec_Whh    = (const float*)d_in[15];
  const float* dec_bih    = (const float*)d_in[16];
  const float* dec_bhh    = (const float*)d_in[17];
  const float* dec_out_W  = (const float*)d_in[18];
  const float* dec_out_b  = (const float*)d_in[19];
  const float* pu_Wih     = (const float*)d_in[20];
  const float* pu_Whh     = (const float*)d_in[21];
  const float* pu_bih     = (const float*)d_in[22];
  const float* pu_bhh     = (const float*)d_in[23];
  const float* pu_qW      = (const float*)d_in[24];
  const float* pu_qb      = (const float*)d_in[25];
  const float* pu_AmapW   = (const float*)d_in[26];
  const float* pu_Amapb   = (const float*)d_in[27];
  const float* pu_qproj   = (const float*)d_in[28];
  const float* pu_kproj   = (const float*)d_in[29];
  const float* pu_vproj   = (const float*)d_in[30];
  const float* pu_inb     = (const float*)d_in[31];
  const float* pu_oproj   = (const float*)d_in[32];
  const float* pu_ob      = (const float*)d_in[33];
  const float* aa_qproj   = (const float*)d_in[34];
  const float* aa_kproj   = (const float*)d_in[35];
  const float* aa_vproj   = (const float*)d_in[36];
  const float* aa_inb     = (const float*)d_in[37];
  const float* aa_oproj   = (const float*)d_in[38];
  const float* aa_ob      = (const float*)d_in[39];
  const float* toh_W      = (const float*)d_in[40];
  const float* toh_b      = (const float*)d_in[41];
  const float* out_W      = (const float*)d_in[42];
  const float* out_b      = (const float*)d_in[43];
  float* out              = (float*)d_out;

  // ---- workspace carve (fp32) ----
  float* p = (float*)d_ws;
  const size_t SBH = (size_t)S_ * B_ * H_;
  float* enc    = p; p += SBH;                       // (S,B,H)  sum of dirs
  float* E2     = p; p += SBH;                       // hoisted enc@W2^T + b
  float* Kc     = p; p += SBH;
  float* Vc     = p; p += SBH;
  float* Kp     = p; p += HA_ * SBH;
  float* Vp     = p; p += HA_ * SBH;
  float* Am     = p; p += (size_t)HA_ * S_ * B_ * ADD_;
  float* h_f    = p; p += (size_t)B_ * H_;           // fwd hidden -> dec_h
  float* h_b    = p; p += (size_t)B_ * H_;
  float* hida   = p; p += (size_t)HA_ * B_ * H_;
  float* ptrs   = p; p += (size_t)HA_ * B_ * ADD_;
  float* gi     = p; p += (size_t)B_ * G_;
  float* gh     = p; p += (size_t)B_ * G_;
  float* qtmp   = p; p += (size_t)B_ * H_;
  float* qv     = p; p += (size_t)B_ * H_;
  float* sc     = p; p += (size_t)B_ * S_;
  float* ws1    = p; p += (size_t)B_ * H_;
  float* cv1cat = p; p += (size_t)B_ * HA_ * H_;
  float* ctx    = p; p += (size_t)B_ * H_;
  float* e1     = p; p += (size_t)B_ * H_;
  float* rnn_in = p; p += (size_t)B_ * DIN_;
  float* cat2   = p; p += (size_t)B_ * 2 * H_;
  float* decout = p; p += (size_t)B_ * H_;
  float* q0     = p; p += (size_t)B_ * H_;
  float* qc     = p; p += (size_t)B_ * H_;
  float* ws2    = p; p += (size_t)B_ * H_;
  float* cv2    = p; p += (size_t)B_ * H_;
  float* cat3   = p; p += (size_t)B_ * (H_ + HA_ * H_ + H_);

  const float scale = 1.0f / sqrtf((float)H_);
  const int BH_BLOCKS = (B_ * H_) / 256;

  // ---- init recurrent state (h_f, h_b, hida are contiguous) ----
  zero_kernel<<<(4 * B_ * H_) / 256, 256, 0, stream>>>(h_f, 4 * B_ * H_);
  init_ptrs_kernel<<<(HA_ * B_ * ADD_ + 255) / 256, 256, 0, stream>>>(Aaddr, ptrs);

  // ================= encoder (bidirectional GRU) =================
  for (int s = 0; s < S_; ++s) {
    gemm(stream, src + (size_t)s * B_ * E_, E_, enc_Wih_f, E_, enc_bih_f, gi, G_, B_, G_, E_);
    gemm(stream, h_f, H_, enc_Whh_f, H_, enc_bhh_f, gh, G_, B_, G_, H_);
    gru_gate_kernel<<<BH_BLOCKS, 256, 0, stream>>>(gi, gh, h_f, enc + (size_t)s * B_ * H_, 0);
  }
  for (int i = 0; i < S_; ++i) {
    const int s = S_ - 1 - i;
    gemm(stream, src + (size_t)s * B_ * E_, E_, enc_Wih_b, E_, enc_bih_b, gi, G_, B_, G_, E_);
    gemm(stream, h_b, H_, enc_Whh_b, H_, enc_bhh_b, gh, G_, B_, G_, H_);
    gru_gate_kernel<<<BH_BLOCKS, 256, 0, stream>>>(gi, gh, h_b, enc + (size_t)s * B_ * H_, 1);
  }
  // h_f now holds hTf == initial decoder hidden; keep using it as dec_h.

  // ================= step-invariant hoists =================
  amap_kernel<<<(HA_ * S_ * B_ * ADD_) / 256, 256, 0, stream>>>(Aaddr, pu_AmapW, pu_Amapb, Am);
  for (int u = 0; u < HA_; ++u) {
    gemm(stream, Am + (size_t)u * S_ * B_ * ADD_, ADD_,
         pu_kproj + (size_t)u * H_ * ADD_, ADD_,
         pu_inb + (size_t)u * 3 * H_ + H_,                 // bk_u
         Kp + (size_t)u * SBH, H_, S_ * B_, H_, ADD_);
    gemm(stream, enc, H_,
         pu_vproj + (size_t)u * H_ * H_, H_,
         pu_inb + (size_t)u * 3 * H_ + 2 * H_,             // bv_u
         Vp + (size_t)u * SBH, H_, S_ * B_, H_, H_);
  }
  gemm(stream, enc, H_, aa_kproj, H_, aa_inb + H_,     Kc, H_, S_ * B_, H_, H_);
  gemm(stream, enc, H_, aa_vproj, H_, aa_inb + 2 * H_, Vc, H_, S_ * B_, H_, H_);
  // E2[s,b,:] = enc[s,b,:] @ W2^T + dec_attn_b  (W2 = cols [H,2H) of dec_attn_W)
  gemm(stream, enc, H_, dec_attn_W + H_, 2 * H_, dec_attn_b, E2, H_, S_ * B_, H_, H_);

  // ================= decoder (T steps) =================
  for (int t = 0; t < T_; ++t) {
    // ---- pointer units ----
    for (int u = 0; u < HA_; ++u) {
      float* hida_u = hida + (size_t)u * B_ * H_;
      gemm(stream, ptrs + (size_t)u * B_ * ADD_, ADD_,
           pu_Wih + (size_t)u * G_ * ADD_, ADD_, pu_bih + (size_t)u * G_,
           gi, G_, B_, G_, ADD_);
      gemm(stream, hida_u, H_,
           pu_Whh + (size_t)u * G_ * H_, H_, pu_bhh + (size_t)u * G_,
           gh, G_, B_, G_, H_);
      gru_gate_kernel<<<BH_BLOCKS, 256, 0, stream>>>(gi, gh, hida_u, nullptr, 0);
      gemm(stream, hida_u, H_, pu_qW + (size_t)u * H_ * H_, H_,
           pu_qb + (size_t)u * H_, qtmp, H_, B_, H_, H_);
      gemm(stream, qtmp, H_, pu_qproj + (size_t)u * H_ * H_, H_,
           pu_inb + (size_t)u * 3 * H_, qv, H_, B_, H_, H_);          // + bq_u
      dot_scores_kernel<<<(B_ * S_) / 8, 256, 0, stream>>>(qv, Kp + (size_t)u * SBH, sc, scale);
      softmax256_kernel<<<B_, 256, 0, stream>>>(sc);
      attn_apply_kernel<<<BH_BLOCKS, 256, 0, stream>>>(sc, Vp + (size_t)u * SBH, ws1);
      gemm(stream, ws1, H_, pu_oproj + (size_t)u * H_ * H_, H_,
           pu_ob + (size_t)u * H_, cv1cat + (size_t)u * H_, HA_ * H_, B_, H_, H_);
      cadd_kernel<<<(B_ * ADD_ + 255) / 256, 256, 0, stream>>>(sc, Aaddr, ptrs + (size_t)u * B_ * ADD_);
    }
    // ---- Bahdanau attention (uses pre-update dec_h == h_f) ----
    gemm(stream, h_f, H_, dec_attn_W, 2 * H_, nullptr, e1, H_, B_, H_, H_);
    bahdanau_scores_kernel<<<(B_ * S_) / 8, 256, 0, stream>>>(e1, E2, dec_attn_v, sc);
    softmax256_kernel<<<B_, 256, 0, stream>>>(sc);
    attn_apply_kernel<<<BH_BLOCKS, 256, 0, stream>>>(sc, enc, ctx);
    // ---- decoder GRU ----
    const float* emb = tgt + (size_t)(t == 0 ? 0 : t - 1) * B_ * E_;
    copy_cols_kernel<<<(B_ * E_) / 256, 256, 0, stream>>>(emb, E_, rnn_in, DIN_, E_);
    copy_cols_kernel<<<(B_ * HA_ * H_) / 256, 256, 0, stream>>>(cv1cat, HA_ * H_, rnn_in + E_, DIN_, HA_ * H_);
    copy_cols_kernel<<<(B_ * H_) / 256, 256, 0, stream>>>(ctx, H_, rnn_in + E_ + HA_ * H_, DIN_, H_);
    gemm(stream, rnn_in, DIN_, dec_Wih, DIN_, dec_bih, gi, G_, B_, G_, DIN_);
    gemm(stream, h_f, H_, dec_Whh, H_, dec_bhh, gh, G_, B_, G_, H_);
    gru_gate_kernel<<<BH_BLOCKS, 256, 0, stream>>>(gi, gh, h_f, nullptr, 0);
    // ---- dec_out = [dec_h2, ctx] @ dec_out_W^T + b ----
    copy_cols_kernel<<<(B_ * H_) / 256, 256, 0, stream>>>(h_f, H_, cat2, 2 * H_, H_);
    copy_cols_kernel<<<(B_ * H_) / 256, 256, 0, stream>>>(ctx, H_, cat2 + H_, 2 * H_, H_);
    gemm(stream, cat2, 2 * H_, dec_out_W, 2 * H_, dec_out_b, decout, H_, B_, H_, 2 * H_);
    // ---- content-based pointer ----
    gemm(stream, cv1cat, HA_ * H_, toh_W, HA_ * H_, toh_b, q0, H_, B_, H_, HA_ * H_);
    gemm(stream, q0, H_, aa_qproj, H_, aa_inb, qc, H_, B_, H_, H_);     // + aa_bq
    dot_scores_kernel<<<(B_ * S_) / 8, 256, 0, stream>>>(qc, Kc, sc, scale);
    softmax256_kernel<<<B_, 256, 0, stream>>>(sc);
    attn_apply_kernel<<<BH_BLOCKS, 256, 0, stream>>>(sc, Vc, ws2);
    gemm(stream, ws2, H_, aa_oproj, H_, aa_ob, cv2, H_, B_, H_, H_);
    // ---- output head -> d_out[t] ----
    copy_cols_kernel<<<(B_ * H_) / 256, 256, 0, stream>>>(decout, H_, cat3, DIN_, H_);
    copy_cols_kernel<<<(B_ * HA_ * H_) / 256, 256, 0, stream>>>(cv1cat, HA_ * H_, cat3 + H_, DIN_, HA_ * H_);
    copy_cols_kernel<<<(B_ * H_) / 256, 256, 0, stream>>>(cv2, H_, cat3 + H_ + HA_ * H_, DIN_, H_);
    gemm(stream, cat3, DIN_, out_W, DIN_, out_b, out + (size_t)t * B_ * OUT_, OUT_, B_, OUT_, DIN_);
  }
}